// Feature_Extractor_Student_53635551593265
// MI455X (gfx1250) — compile-verified
//
#include <hip/hip_runtime.h>
#include <cstddef>

// ---------------------------------------------------------------------------
// DGCNN-style feature extractor for MI455X (gfx1250, wave32, WMMA).
// All matrix math goes through V_WMMA_F32_16X16X4_F32 (fp32 matrix path).
// Stage shapes are compile-time template parameters so the hot loops have
// no bounds guards (branchless clamp+select only for the ragged C=3 stage).
// ---------------------------------------------------------------------------

#define BB 4
#define NN 2048
#define BN (BB * NN)
#define KNN 20
#define NEG_INF (-3.4e38f)

typedef __attribute__((ext_vector_type(2))) float v2f;
typedef __attribute__((ext_vector_type(8))) float v8f;

__device__ __forceinline__ v8f wmma4(v2f a, v2f b, v8f c) {
  // D = A(16x4,f32) * B(4x16,f32) + C(16x16,f32)
  return __builtin_amdgcn_wmma_f32_16x16x4_f32(false, a, false, b, (short)0, c,
                                               false, false);
}

__device__ __forceinline__ v8f vzero8() {
  v8f z = {0.f, 0.f, 0.f, 0.f, 0.f, 0.f, 0.f, 0.f};
  return z;
}

// Branchless K-fragment load: for K multiple of 4 -> one unconditional b64
// load; otherwise clamped-index unconditional loads + cndmask zeroing
// (no EXEC-mask branching around VMEM).
template <int K>
__device__ __forceinline__ v2f load_k2(const float* __restrict__ p, int c0) {
  v2f r;
  if ((K & 3) == 0) {
    r = *(const v2f*)(p + c0);
  } else {
    int i0 = (c0 < K) ? c0 : 0;
    int i1 = (c0 + 1 < K) ? c0 + 1 : 0;
    float a = p[i0];
    float b = p[i1];
    r.x = (c0 < K) ? a : 0.f;
    r.y = (c0 + 1 < K) ? b : 0.f;
  }
  return r;
}

// ---------------------------------------------------------------------------
// squared norms per point
// ---------------------------------------------------------------------------
template <int C>
__global__ void sqnorm_kernel(const float* __restrict__ x,
                              float* __restrict__ xx) {
  int i = blockIdx.x * blockDim.x + threadIdx.x;
  if (i >= BN) return;
  const float* p = x + (size_t)i * C;
  float s = 0.f;
#pragma unroll
  for (int c = 0; c < C; ++c) s += p[c] * p[c];
  xx[i] = s;
}

// ---------------------------------------------------------------------------
// Fused kNN: Gram tile via WMMA -> neg-dist scores in LDS -> top-20 per row.
// One wave per (batch, 16-row tile).
// ---------------------------------------------------------------------------
template <int C>
__global__ __launch_bounds__(32) void knn_topk_kernel(
    const float* __restrict__ x, const float* __restrict__ xx,
    int* __restrict__ idxOut) {
  constexpr int CHUNKS = (C + 3) >> 2;
  constexpr int CP = CHUNKS * 4;
  const int l = threadIdx.x;
  const int lm = l & 15, lh = l >> 4;
  const int b = blockIdx.x >> 7;            // N/16 = 128 row tiles per batch
  const int row0 = (blockIdx.x & 127) * 16;

  __shared__ float sA[16 * 256];            // row tile, zero padded K
  __shared__ float sxx[16];
  __shared__ float sS[16 * 512];            // score slab: 16 rows x 512 cols

  const float* xb = x + (size_t)b * NN * C;
  for (int t = l; t < 16 * CP; t += 32) {
    int r = t / CP, c = t - r * CP;
    sA[r * 256 + c] = (c < C) ? xb[(size_t)(row0 + r) * C + c] : 0.f;
  }
  if (l < 16) sxx[l] = xx[b * NN + row0 + l];
  __syncthreads();

  float vals[KNN];
  int ids[KNN];
#pragma unroll
  for (int t = 0; t < KNN; ++t) { vals[t] = NEG_INF; ids[t] = 0; }

  for (int cc = 0; cc < 4; ++cc) {          // 4 chunks of 512 columns
    for (int ct = 0; ct < 32; ++ct) {
      const int col0 = cc * 512 + ct * 16;
      const int col = col0 + lm;
      const float xxc = xx[b * NN + col];
      const float* bp = xb + (size_t)col * C;
      v8f acc = vzero8();
#pragma unroll 4
      for (int kc = 0; kc < CHUNKS; ++kc) {
        const int c0 = kc * 4 + lh * 2;
        v2f af = *(const v2f*)(&sA[lm * 256 + c0]);   // ds_load_b64
        v2f bf = load_k2<C>(bp, c0);                  // global_load_b64
        acc = wmma4(af, bf, acc);
      }
#pragma unroll
      for (int v = 0; v < 8; ++v) {
        const int M = v + 8 * lh;
        sS[M * 512 + ct * 16 + lm] = 2.f * acc[v] - sxx[M] - xxc;
      }
    }
    __syncthreads();
    if (l < 16) {
      for (int j = 0; j < 512; ++j) {
        float s = sS[l * 512 + j];
        if (s > vals[KNN - 1]) {
          float cv = s; int ci = cc * 512 + j;
#pragma unroll
          for (int t = 0; t < KNN; ++t) {
            if (cv > vals[t]) {
              float tv = vals[t]; int ti = ids[t];
              vals[t] = cv; ids[t] = ci; cv = tv; ci = ti;
            }
          }
        }
      }
    }
    __syncthreads();
  }
  if (l < 16) {
    int* op = idxOut + (size_t)(b * NN + row0 + l) * KNN;
#pragma unroll
    for (int t = 0; t < KNN; ++t) op[t] = ids[t];
  }
}

// ---------------------------------------------------------------------------
// fold eval-BN into scale/shift
// ---------------------------------------------------------------------------
__global__ void bnprep_kernel(const float* g, const float* bb, const float* m,
                              const float* v, int O, float* sc, float* sh) {
  int o = blockIdx.x * blockDim.x + threadIdx.x;
  if (o >= O) return;
  float s = g[o] * rsqrtf(v[o] + 1e-5f);
  sc[o] = s;
  sh[o] = bb[o] - m[o] * s;
}

// wdiff[o][c] = w[o][C+c] - w[o][c]   (self-term weights)
__global__ void wdiff_kernel(const float* __restrict__ w, int C, int O,
                             float* __restrict__ wd) {
  int i = blockIdx.x * blockDim.x + threadIdx.x;
  if (i >= O * C) return;
  int o = i / C, c = i - o * C;
  wd[i] = w[(size_t)o * 2 * C + C + c] - w[(size_t)o * 2 * C + c];
}

// ---------------------------------------------------------------------------
// NT GEMM (out[M,N] = A[M,K] * B[N,K]^T) via f32 WMMA. Wave = 16x64 tile.
// M multiple of 16, N multiple of 64; K templated (ragged K is branchless).
// ---------------------------------------------------------------------------
template <int K>
__global__ __launch_bounds__(32) void gemm_nt_kernel(
    const float* __restrict__ A, const float* __restrict__ Bm, int Nn,
    float* __restrict__ out) {
  constexpr int CHUNKS = (K + 3) >> 2;
  const int nt4 = Nn >> 6;
  const int m0 = (blockIdx.x / nt4) * 16;
  const int n0 = (blockIdx.x % nt4) * 64;
  const int l = threadIdx.x, lm = l & 15, lh = l >> 4;
  const float* arow = A + (size_t)(m0 + lm) * K;
  const float* brow0 = Bm + (size_t)(n0 + lm) * K;

  v8f acc[4];
#pragma unroll
  for (int j = 0; j < 4; ++j) acc[j] = vzero8();

#pragma unroll 4
  for (int kc = 0; kc < CHUNKS; ++kc) {
    const int c0 = kc * 4 + lh * 2;
    v2f af = load_k2<K>(arow, c0);
#pragma unroll
    for (int j = 0; j < 4; ++j) {
      v2f bf = load_k2<K>(brow0 + (size_t)(j * 16) * K, c0);
      acc[j] = wmma4(af, bf, acc[j]);
    }
  }
#pragma unroll
  for (int j = 0; j < 4; ++j) {
    const int nc = n0 + j * 16 + lm;
#pragma unroll
    for (int v = 0; v < 8; ++v) {
      const int row = m0 + v + 8 * lh;
      out[(size_t)row * Nn + nc] = acc[j][v];
    }
  }
}

// ---------------------------------------------------------------------------
// EdgeConv: per point, gathered-neighbor GEMM (two 16-row k-tiles, k=20
// padded with neighbor 19) via WMMA + self term + BN + LeakyReLU + max_k.
// One wave per (b,n); loops over output-channel tiles of 16.
// ---------------------------------------------------------------------------
template <int C, int O>
__global__ __launch_bounds__(32) void edgeconv_kernel(
    const float* __restrict__ x, const int* __restrict__ idx,
    const float* __restrict__ w, const float* __restrict__ T,
    const float* __restrict__ bsc, const float* __restrict__ bsh,
    float* __restrict__ out) {
  constexpr int CHUNKS = (C + 3) >> 2;
  constexpr int twoC = 2 * C;
  const int bn = blockIdx.x;
  const int b = bn >> 11;
  const int l = threadIdx.x, lm = l & 15, lh = l >> 4;
  const int* ip = idx + (size_t)bn * KNN;
  int k1 = 16 + lm; if (k1 > KNN - 1) k1 = KNN - 1;   // pad rows dup k=19
  const float* rowA0 = x + ((size_t)(b * NN) + ip[lm]) * C;
  const float* rowA1 = x + ((size_t)(b * NN) + ip[k1]) * C;

  for (int o0 = 0; o0 < O; o0 += 16) {
    const float* wrow = w + (size_t)(o0 + lm) * twoC;  // w_a part: cols [0,C)
    v8f acc0 = vzero8(), acc1 = vzero8();
#pragma unroll 4
    for (int kc = 0; kc < CHUNKS; ++kc) {
      const int c0 = kc * 4 + lh * 2;
      v2f bf = load_k2<C>(wrow, c0);     // twoC stride is 8-float aligned
      v2f a0 = load_k2<C>(rowA0, c0);
      v2f a1 = load_k2<C>(rowA1, c0);
      acc0 = wmma4(a0, bf, acc0);        // independent accs -> co-execution
      acc1 = wmma4(a1, bf, acc1);
    }
    const int col = o0 + lm;
    const float t = T[(size_t)bn * O + col];
    const float sc = bsc[col], sh = bsh[col];
    float mx = NEG_INF;
#pragma unroll
    for (int v = 0; v < 8; ++v) {
      float h0 = (acc0[v] + t) * sc + sh;
      h0 = h0 >= 0.f ? h0 : 0.2f * h0;
      float h1 = (acc1[v] + t) * sc + sh;
      h1 = h1 >= 0.f ? h1 : 0.2f * h1;
      mx = fmaxf(mx, fmaxf(h0, h1));
    }
    mx = fmaxf(mx, __shfl_xor(mx, 16, 32));  // combine M halves of C layout
    if (l < 16) out[(size_t)bn * O + col] = mx;
  }
}

// ---------------------------------------------------------------------------
// NN GEMM (out[M,N] = A[M,K] * B[K,N]) via f32 WMMA, fused bias/relu/residual.
// K multiple of 4; M multiple of 16; N multiple of 64.
// ---------------------------------------------------------------------------
template <int K, int RELU, int RES>
__global__ __launch_bounds__(32) void gemm_nn_kernel(
    const float* __restrict__ A, const float* __restrict__ Bk, int Nn,
    const float* __restrict__ bias, const float* __restrict__ res,
    float* __restrict__ out) {
  const int nt4 = Nn >> 6;
  const int m0 = (blockIdx.x / nt4) * 16;
  const int n0 = (blockIdx.x % nt4) * 64;
  const int l = threadIdx.x, lm = l & 15, lh = l >> 4;
  const float* arow = A + (size_t)(m0 + lm) * K;

  v8f acc[4];
#pragma unroll
  for (int j = 0; j < 4; ++j) acc[j] = vzero8();

#pragma unroll 4
  for (int kc = 0; kc < K / 4; ++kc) {
    const int c0 = kc * 4 + lh * 2;
    v2f af = *(const v2f*)(arow + c0);   // global_load_b64
#pragma unroll
    for (int j = 0; j < 4; ++j) {
      const int nc = n0 + j * 16 + lm;
      v2f bf;
      bf.x = Bk[(size_t)c0 * Nn + nc];
      bf.y = Bk[(size_t)(c0 + 1) * Nn + nc];
      acc[j] = wmma4(af, bf, acc[j]);
    }
  }
#pragma unroll
  for (int j = 0; j < 4; ++j) {
    const int nc = n0 + j * 16 + lm;
    const float bv = bias[nc];
#pragma unroll
    for (int v = 0; v < 8; ++v) {
      const int row = m0 + v + 8 * lh;
      float val = acc[j][v] + bv;
      if (RES) val += res[(size_t)row * Nn + nc];
      if (RELU) val = fmaxf(val, 0.f);
      out[(size_t)row * Nn + nc] = val;
    }
  }
}

// (B,N,C) -> (B,C,N)
__global__ void transpose_kernel(const float* __restrict__ in, int C,
                                 float* __restrict__ outp) {
  size_t i = (size_t)blockIdx.x * blockDim.x + threadIdx.x;
  size_t tot = (size_t)BB * NN * C;
  if (i >= tot) return;
  int n = (int)(i % NN);
  size_t r = i / NN;
  int c = (int)(r % C);
  int b = (int)(r / C);
  outp[i] = in[((size_t)b * NN + n) * C + c];
}

__global__ void qkv_kernel(const float* __restrict__ x3, const float* wq,
                           const float* bq, const float* wk, const float* bk,
                           const float* wv, const float* bv, float* q, float* k,
                           float* v) {
  int i = blockIdx.x * blockDim.x + threadIdx.x;
  if (i >= BN) return;
  const float* p = x3 + (size_t)i * 1024;
  float sq = 0.f, sk = 0.f, sv = 0.f;
  for (int c = 0; c < 1024; ++c) {
    float xv = p[c];
    sq += xv * wq[c]; sk += xv * wk[c]; sv += xv * wv[c];
  }
  q[i] = sq + bq[0]; k[i] = sk + bk[0]; v[i] = sv + bv[0];
}

__global__ void kstats_kernel(const float* __restrict__ k, float* kmx,
                              float* kmn) {
  __shared__ float smx[256], smn[256];
  int b = blockIdx.x, t = threadIdx.x;
  float mx = NEG_INF, mn = 3.4e38f;
  for (int n = t; n < NN; n += 256) {
    float kv = k[b * NN + n];
    mx = fmaxf(mx, kv); mn = fminf(mn, kv);
  }
  smx[t] = mx; smn[t] = mn;
  __syncthreads();
  for (int s = 128; s > 0; s >>= 1) {
    if (t < s) { smx[t] = fmaxf(smx[t], smx[t + s]); smn[t] = fminf(smn[t], smn[t + s]); }
    __syncthreads();
  }
  if (t == 0) { kmx[b] = smx[0]; kmn[b] = smn[0]; }
}

// scores[n,m] = q[n]*k[m]; softmax max over m is q*kmax (q>=0) or q*kmin.
__global__ void attn_kernel(const float* __restrict__ q,
                            const float* __restrict__ k,
                            const float* __restrict__ v, const float* kmx,
                            const float* kmn, float* __restrict__ att) {
  int bn = blockIdx.x, b = bn >> 11, t = threadIdx.x;
  float qv = q[bn];
  float mx = qv >= 0.f ? qv * kmx[b] : qv * kmn[b];
  float se = 0.f, sv = 0.f;
  for (int m = t; m < NN; m += 256) {
    float e = expf(qv * k[b * NN + m] - mx);
    se += e; sv += e * v[b * NN + m];
  }
  __shared__ float s1[256], s2[256];
  s1[t] = se; s2[t] = sv;
  __syncthreads();
  for (int s = 128; s > 0; s >>= 1) {
    if (t < s) { s1[t] += s1[t + s]; s2[t] += s2[t + s]; }
    __syncthreads();
  }
  if (t == 0) att[bn] = s2[0] / s1[0];
}

// apre = att * wo + bo + x3 (residual)
__global__ void apre_kernel(const float* __restrict__ att,
                            const float* __restrict__ wo,
                            const float* __restrict__ bo,
                            const float* __restrict__ x3,
                            float* __restrict__ out) {
  size_t i = (size_t)blockIdx.x * blockDim.x + threadIdx.x;
  if (i >= (size_t)BN * 1024) return;
  int o = (int)(i & 1023);
  size_t bn = i >> 10;
  out[i] = att[bn] * wo[o] + bo[o] + x3[i];
}

__global__ void layernorm_kernel(const float* __restrict__ in,
                                 const float* __restrict__ g,
                                 const float* __restrict__ bt,
                                 float* __restrict__ out) {
  int bn = blockIdx.x, t = threadIdx.x;
  const float* p = in + (size_t)bn * 1024;
  float vals[4];
  float s = 0.f, s2 = 0.f;
#pragma unroll
  for (int j = 0; j < 4; ++j) {
    float xv = p[t + 256 * j];
    vals[j] = xv; s += xv; s2 += xv * xv;
  }
  __shared__ float ss[256], ssq[256];
  ss[t] = s; ssq[t] = s2;
  __syncthreads();
  for (int st = 128; st > 0; st >>= 1) {
    if (t < st) { ss[t] += ss[t + st]; ssq[t] += ssq[t + st]; }
    __syncthreads();
  }
  float mu = ss[0] * (1.f / 1024.f);
  float var = ssq[0] * (1.f / 1024.f) - mu * mu;
  float inv = rsqrtf(var + 1e-5f);
#pragma unroll
  for (int j = 0; j < 4; ++j) {
    int o = t + 256 * j;
    out[(size_t)bn * 1024 + o] = (vals[j] - mu) * inv * g[o] + bt[o];
  }
}

__global__ void maxpool_kernel(const float* __restrict__ a2,
                               float* __restrict__ outp) {
  int i = blockIdx.x * blockDim.x + threadIdx.x;
  if (i >= BB * 1024) return;
  int b = i >> 10, o = i & 1023;
  const float* p = a2 + (size_t)b * NN * 1024 + o;
  float mx = NEG_INF;
  for (int n = 0; n < NN; ++n) mx = fmaxf(mx, p[(size_t)n * 1024]);
  outp[i] = mx;  // (B,1024,1) row-major
}

// ---------------------------------------------------------------------------
// Host launcher
// ---------------------------------------------------------------------------
extern "C" void kernel_launch(void* const* d_in, const int* in_sizes, int n_in,
                              void* d_out, int out_size, void* d_ws,
                              size_t ws_size, hipStream_t stream) {
  (void)in_sizes; (void)n_in; (void)out_size; (void)ws_size;
  const float* x    = (const float*)d_in[0];
  const float* w1   = (const float*)d_in[1];
  const float* g1   = (const float*)d_in[2];
  const float* b1   = (const float*)d_in[3];
  const float* m1   = (const float*)d_in[4];
  const float* v1   = (const float*)d_in[5];
  const float* w2   = (const float*)d_in[6];
  const float* g2   = (const float*)d_in[7];
  const float* b2   = (const float*)d_in[8];
  const float* m2   = (const float*)d_in[9];
  const float* v2   = (const float*)d_in[10];
  const float* w3   = (const float*)d_in[11];
  const float* g3   = (const float*)d_in[12];
  const float* b3   = (const float*)d_in[13];
  const float* m3   = (const float*)d_in[14];
  const float* v3   = (const float*)d_in[15];
  const float* wq   = (const float*)d_in[16];
  const float* bq   = (const float*)d_in[17];
  const float* wk   = (const float*)d_in[18];
  const float* bk   = (const float*)d_in[19];
  const float* wv   = (const float*)d_in[20];
  const float* bv   = (const float*)d_in[21];
  const float* wo   = (const float*)d_in[22];
  const float* bo   = (const float*)d_in[23];
  const float* fw1  = (const float*)d_in[24];
  const float* fb1  = (const float*)d_in[25];
  const float* fw2  = (const float*)d_in[26];
  const float* fb2  = (const float*)d_in[27];
  const float* ln1g = (const float*)d_in[28];
  const float* ln1b = (const float*)d_in[29];
  const float* ln2g = (const float*)d_in[30];
  const float* ln2b = (const float*)d_in[31];
  float* outp = (float*)d_out;

  // workspace carve-up
  size_t off = 0;
  auto take = [&](size_t bytes) -> void* {
    void* p = (char*)d_ws + off;
    off += (bytes + 255) & ~(size_t)255;
    return p;
  };
  int*   idxb = (int*)  take((size_t)BN * KNN * 4);
  float* xx   = (float*)take((size_t)BN * 4);
  float* x1   = (float*)take((size_t)BN * 128 * 4);
  float* x2   = (float*)take((size_t)BN * 256 * 4);
  float* x3   = (float*)take((size_t)BN * 1024 * 4);
  float* Tbuf = (float*)take((size_t)BN * 1024 * 4);
  float* wd   = (float*)take((size_t)1024 * 256 * 4);
  float* bsc  = (float*)take(1024 * 4);
  float* bsh  = (float*)take(1024 * 4);
  float* qb   = (float*)take((size_t)BN * 4);
  float* kb   = (float*)take((size_t)BN * 4);
  float* vb   = (float*)take((size_t)BN * 4);
  float* attb = (float*)take((size_t)BN * 4);
  float* kmx  = (float*)take(BB * 4);
  float* kmn  = (float*)take(BB * 4);
  float* apre = (float*)take((size_t)BN * 1024 * 4);
  float* a1   = (float*)take((size_t)BN * 1024 * 4);
  float* hbuf = (float*)take((size_t)BN * 2048 * 4);
  float* fres = apre;  // reuse: apre consumed by LN1 before FFN2 writes
  float* a2   = Tbuf;  // reuse: T consumed before LN2

  // ---- stage 1: C=3 -> O=128 -------------------------------------------
  sqnorm_kernel<3><<<(BN + 255) / 256, 256, 0, stream>>>(x, xx);
  knn_topk_kernel<3><<<BB * (NN / 16), 32, 0, stream>>>(x, xx, idxb);
  bnprep_kernel<<<1, 256, 0, stream>>>(g1, b1, m1, v1, 128, bsc, bsh);
  wdiff_kernel<<<(128 * 3 + 255) / 256, 256, 0, stream>>>(w1, 3, 128, wd);
  gemm_nt_kernel<3><<<(BN / 16) * (128 / 64), 32, 0, stream>>>(x, wd, 128,
                                                               Tbuf);
  edgeconv_kernel<3, 128><<<BN, 32, 0, stream>>>(x, idxb, w1, Tbuf, bsc, bsh,
                                                 x1);

  // ---- stage 2: C=128 -> O=256 -----------------------------------------
  sqnorm_kernel<128><<<(BN + 255) / 256, 256, 0, stream>>>(x1, xx);
  knn_topk_kernel<128><<<BB * (NN / 16), 32, 0, stream>>>(x1, xx, idxb);
  bnprep_kernel<<<1, 256, 0, stream>>>(g2, b2, m2, v2, 256, bsc, bsh);
  wdiff_kernel<<<(256 * 128 + 255) / 256, 256, 0, stream>>>(w2, 128, 256, wd);
  gemm_nt_kernel<128><<<(BN / 16) * (256 / 64), 32, 0, stream>>>(x1, wd, 256,
                                                                 Tbuf);
  edgeconv_kernel<128, 256><<<BN, 32, 0, stream>>>(x1, idxb, w2, Tbuf, bsc,
                                                   bsh, x2);

  // ---- stage 3: C=256 -> O=1024 ----------------------------------------
  sqnorm_kernel<256><<<(BN + 255) / 256, 256, 0, stream>>>(x2, xx);
  knn_topk_kernel<256><<<BB * (NN / 16), 32, 0, stream>>>(x2, xx, idxb);
  bnprep_kernel<<<4, 256, 0, stream>>>(g3, b3, m3, v3, 1024, bsc, bsh);
  wdiff_kernel<<<(1024 * 256 + 255) / 256, 256, 0, stream>>>(w3, 256, 1024,
                                                             wd);
  gemm_nt_kernel<256><<<(BN / 16) * (1024 / 64), 32, 0, stream>>>(x2, wd, 1024,
                                                                  Tbuf);
  edgeconv_kernel<256, 1024><<<BN, 32, 0, stream>>>(x2, idxb, w3, Tbuf, bsc,
                                                    bsh, x3);

  // tuple outputs 2,3: x1^T (B,128,N), x2^T (B,256,N)
  {
    size_t e1 = (size_t)BB * 128 * NN;
    size_t e2 = (size_t)BB * 256 * NN;
    transpose_kernel<<<(unsigned)((e1 + 255) / 256), 256, 0, stream>>>(
        x1, 128, outp + BB * 1024);
    transpose_kernel<<<(unsigned)((e2 + 255) / 256), 256, 0, stream>>>(
        x2, 256, outp + BB * 1024 + e1);
  }

  // attention (q=k=v dim 1) + projection + residual + LN1
  qkv_kernel<<<(BN + 255) / 256, 256, 0, stream>>>(x3, wq, bq, wk, bk, wv, bv,
                                                   qb, kb, vb);
  kstats_kernel<<<BB, 256, 0, stream>>>(kb, kmx, kmn);
  attn_kernel<<<BN, 256, 0, stream>>>(qb, kb, vb, kmx, kmn, attb);
  apre_kernel<<<(unsigned)(((size_t)BN * 1024 + 255) / 256), 256, 0, stream>>>(
      attb, wo, bo, x3, apre);
  layernorm_kernel<<<BN, 256, 0, stream>>>(apre, ln1g, ln1b, a1);

  // positionwise FFN (both GEMMs via WMMA) + residual + LN2
  gemm_nn_kernel<1024, 1, 0><<<(BN / 16) * (2048 / 64), 32, 0, stream>>>(
      a1, fw1, 2048, fb1, nullptr, hbuf);
  gemm_nn_kernel<2048, 0, 1><<<(BN / 16) * (1024 / 64), 32, 0, stream>>>(
      hbuf, fw2, 1024, fb2, a1, fres);
  layernorm_kernel<<<BN, 256, 0, stream>>>(fres, ln2g, ln2b, a2);

  // tuple output 1: max over N -> (B,1024,1)
  maxpool_kernel<<<(BB * 1024 + 255) / 256, 256, 0, stream>>>(a2, outp);
}